// NodeUpdateBlock_6897717477397
// MI455X (gfx1250) — compile-verified
//
#include <hip/hip_runtime.h>
#include <hip/hip_bf16.h>
#include <math.h>

// ---------------- problem constants ----------------
#define NN     15000
#define EE     120000
#define NS_    16
#define NV_    8
#define ES_    8
#define EV_    4
#define U0_    40      // NS+NS+ES
#define U1_    20      // NV+NV+EV
#define OS_    16
#define OG_    8
#define OV_    8
#define STOT   24      // OS+OG
#define FCIN   64
#define FCH    16

#define RS3f     0.5773502691896258f   // 1/sqrt(3)
#define RSQ40f   0.15811388300841897f  // 1/sqrt(40)
#define RSQ20f   0.22360679774997896f  // 1/sqrt(20)
#define RSQ24f   0.20412414523193154f  // 1/sqrt(24)
#define RSQ8f    0.35355339059327373f  // 1/sqrt(8)
#define RSQ128f  0.08838834764831845f  // 1/sqrt(128)

// ---------------- workspace layout (floats) ----------------
#define OFF_SPRE 0            // N*16
#define OFF_VPRE 240000       // N*24
#define OFF_AGGS 600000       // N*24
#define OFF_AGGV 960000       // N*24
#define OFF_BA   1320000      // 160 ksteps * 2 tiles * 32 lanes * 2 = 20480
#define OFF_BB   1340480      // 80*2*32*2 = 10240
#define OFF_BC   1350720      // 80*1*32*2 = 5120
#define OFF_STAT 1355840      // 64

typedef float v2f __attribute__((ext_vector_type(2)));
typedef float v8f __attribute__((ext_vector_type(8)));

static __device__ __forceinline__ float silu_f(float x) {
    return x / (1.0f + __expf(-x));
}
static __device__ __forceinline__ float sigmoid_f(float x) {
    return 1.0f / (1.0f + __expf(-x));
}

// =====================================================================
// K0: permute fc_w2 into WMMA-B-fragment order for the three GEMM groups
//     (scales folded in), and zero the BN stats buffer.
// B-fragment layout assumption (f32 16x16x4): element B[k,n] lives at
// lane = n + 16*(k>=2), vgpr r = k%2.  Buffer addr = ((kstep*T+tile)*32+lane)*2+r.
// =====================================================================
__global__ __launch_bounds__(256) void k0_prep(const float* __restrict__ fc_w2,
                                               float* __restrict__ BA,
                                               float* __restrict__ BB,
                                               float* __restrict__ BC,
                                               float* __restrict__ stats) {
    int idx = blockIdx.x * blockDim.x + threadIdx.x;
    if (idx < 20480) {  // G_A : K=640 (kh*40+u), 32 cols
        int r = idx & 1, lane = (idx >> 1) & 31, tile = (idx >> 6) & 1, kstep = idx >> 7;
        int k = kstep * 4 + r + 2 * (lane >> 4);
        int col = tile * 16 + (lane & 15);
        int kh = k / 40, u = k % 40;
        float v;
        if (col < 24) v = fc_w2[kh * 1920 + u * 24 + col];              // w1 region
        else          v = fc_w2[kh * 1920 + 1440 + u * 8 + (col - 24)]; // w3 region
        BA[idx] = v * (0.25f * RSQ40f);
    }
    int j = idx - 20480;
    if (j >= 0 && j < 10240) {  // G_B : K=320 (kh*20+u), cols 0-23 valid
        int r = j & 1, lane = (j >> 1) & 31, tile = (j >> 6) & 1, kstep = j >> 7;
        int k = kstep * 4 + r + 2 * (lane >> 4);
        int col = tile * 16 + (lane & 15);
        int kh = k / 20, u = k % 20;
        float v = 0.0f;
        if (col < 24) v = fc_w2[kh * 1920 + 960 + u * 24 + col] * (0.25f * RSQ20f); // w2
        BB[j] = v;
    }
    int q = idx - 30720;
    if (q >= 0 && q < 5120) {  // G_C : K=320, cols 0-7 valid (shared across xyz)
        int r = q & 1, lane = (q >> 1) & 31, kstep = q >> 6;
        int k = kstep * 4 + r + 2 * (lane >> 4);
        int col = lane & 15;
        int kh = k / 20, u = k % 20;
        float v = 0.0f;
        if (col < 8) v = fc_w2[kh * 1920 + 1760 + u * 8 + col] * (0.25f * RSQ20f);  // w4
        BC[q] = v;
    }
    if (idx >= 35840 && idx < 35904) stats[idx - 35840] = 0.0f;
}

// =====================================================================
// K1: per-node pre-linear + self-connection; writes s_pre, v_pre and
//     initializes agg_s/agg_v with sc_s/sc_v so the edge scatter adds on top.
// =====================================================================
__global__ __launch_bounds__(256) void k1_node(const float* __restrict__ node_fea,
                                               const float* __restrict__ Wps,
                                               const float* __restrict__ Wpv,
                                               const float* __restrict__ Wss,
                                               const float* __restrict__ Wvv,
                                               const float* __restrict__ Wsv,
                                               const float* __restrict__ Wvs,
                                               float* __restrict__ s_pre,
                                               float* __restrict__ v_pre,
                                               float* __restrict__ agg_s,
                                               float* __restrict__ agg_v) {
    __shared__ float sW[10048];
    float* sWps = sW;           // 256
    float* sWpv = sW + 256;     // 64
    float* sWss = sW + 320;     // 6144
    float* sWvv = sW + 6464;    // 1536
    float* sWsv = sW + 8000;    // 1024
    float* sWvs = sW + 9024;    // 1024
    for (int t = threadIdx.x; t < 256;  t += 256) sWps[t] = Wps[t];
    for (int t = threadIdx.x; t < 64;   t += 256) sWpv[t] = Wpv[t];
    for (int t = threadIdx.x; t < 6144; t += 256) sWss[t] = Wss[t];
    for (int t = threadIdx.x; t < 1536; t += 256) sWvv[t] = Wvv[t];
    for (int t = threadIdx.x; t < 1024; t += 256) sWsv[t] = Wsv[t];
    for (int t = threadIdx.x; t < 1024; t += 256) sWvs[t] = Wvs[t];
    __syncthreads();

    int nid = blockIdx.x * 256 + threadIdx.x;
    if (nid >= NN) return;
    const float* nf = node_fea + (size_t)nid * 40;

    float sp[16];
#pragma unroll
    for (int w = 0; w < 16; w++) sp[w] = 0.0f;
    for (int u = 0; u < 16; u++) {
        float x = nf[u];
        const float* wr = &sWps[u * 16];
#pragma unroll
        for (int w = 0; w < 16; w++) sp[w] += x * wr[w];
    }
#pragma unroll
    for (int w = 0; w < 16; w++) sp[w] *= 0.25f;

    float vp[8][3];
#pragma unroll
    for (int w = 0; w < 8; w++) { vp[w][0] = 0; vp[w][1] = 0; vp[w][2] = 0; }
    for (int u = 0; u < 8; u++) {
        float x0 = nf[16 + u * 3], x1 = nf[16 + u * 3 + 1], x2 = nf[16 + u * 3 + 2];
        const float* wr = &sWpv[u * 8];
#pragma unroll
        for (int w = 0; w < 8; w++) {
            float ww = wr[w];
            vp[w][0] += x0 * ww; vp[w][1] += x1 * ww; vp[w][2] += x2 * ww;
        }
    }
#pragma unroll
    for (int w = 0; w < 8; w++) { vp[w][0] *= RSQ8f; vp[w][1] *= RSQ8f; vp[w][2] *= RSQ8f; }

    float scs[24];
#pragma unroll
    for (int w = 0; w < 24; w++) scs[w] = 0.0f;
    for (int u = 0; u < 16; u++) {
        float su = sp[u] * (1.0f / 16.0f);
        for (int v2i = 0; v2i < 16; v2i++) {
            float p = su * sp[v2i];
            const float* wr = &sWss[(u * 16 + v2i) * 24];
#pragma unroll
            for (int w = 0; w < 24; w++) scs[w] += p * wr[w];
        }
    }
    for (int u = 0; u < 8; u++) {
        for (int v2i = 0; v2i < 8; v2i++) {
            float d = (vp[u][0] * vp[v2i][0] + vp[u][1] * vp[v2i][1] + vp[u][2] * vp[v2i][2]) *
                      (RS3f / 8.0f);
            const float* wr = &sWvv[(u * 8 + v2i) * 24];
#pragma unroll
            for (int w = 0; w < 24; w++) scs[w] += d * wr[w];
        }
    }
    float scv[8][3];
#pragma unroll
    for (int w = 0; w < 8; w++) { scv[w][0] = 0; scv[w][1] = 0; scv[w][2] = 0; }
    for (int u = 0; u < 16; u++) {
        float su = sp[u] * RSQ128f;
        for (int v2i = 0; v2i < 8; v2i++) {
            const float* wr = &sWsv[(u * 8 + v2i) * 8];
            float p0 = su * vp[v2i][0], p1 = su * vp[v2i][1], p2 = su * vp[v2i][2];
#pragma unroll
            for (int w = 0; w < 8; w++) {
                float ww = wr[w];
                scv[w][0] += p0 * ww; scv[w][1] += p1 * ww; scv[w][2] += p2 * ww;
            }
        }
    }
    for (int u = 0; u < 8; u++) {
        for (int v2i = 0; v2i < 16; v2i++) {
            float sv = sp[v2i] * RSQ128f;
            const float* wr = &sWvs[(u * 16 + v2i) * 8];
            float p0 = vp[u][0] * sv, p1 = vp[u][1] * sv, p2 = vp[u][2] * sv;
#pragma unroll
            for (int w = 0; w < 8; w++) {
                float ww = wr[w];
                scv[w][0] += p0 * ww; scv[w][1] += p1 * ww; scv[w][2] += p2 * ww;
            }
        }
    }
#pragma unroll
    for (int w = 0; w < 16; w++) s_pre[(size_t)nid * 16 + w] = sp[w];
#pragma unroll
    for (int w = 0; w < 8; w++) {
        v_pre[(size_t)nid * 24 + w * 3 + 0] = vp[w][0];
        v_pre[(size_t)nid * 24 + w * 3 + 1] = vp[w][1];
        v_pre[(size_t)nid * 24 + w * 3 + 2] = vp[w][2];
    }
#pragma unroll
    for (int w = 0; w < 24; w++) agg_s[(size_t)nid * 24 + w] = scs[w];
#pragma unroll
    for (int w = 0; w < 8; w++) {
        agg_v[(size_t)nid * 24 + w * 3 + 0] = scv[w][0];
        agg_v[(size_t)nid * 24 + w * 3 + 1] = scv[w][1];
        agg_v[(size_t)nid * 24 + w * 3 + 2] = scv[w][2];
    }
}

// =====================================================================
// K_edge: one wave = 16 edges. Builds h, s1, b, v1 in LDS, then runs
// f32 16x16x4 WMMA GEMMs against pre-permuted fc_w2 and scatters with
// global f32 atomics into agg_s/agg_v.
// =====================================================================
#define WMMA4(a, b, c) \
    __builtin_amdgcn_wmma_f32_16x16x4_f32(false, (a), false, (b), (short)0, (c), false, false)

__global__ __launch_bounds__(128) void k_edge(const float* __restrict__ s_pre,
                                              const float* __restrict__ v_pre,
                                              const float* __restrict__ edge_sh,
                                              const float* __restrict__ edge_fea,
                                              const float* __restrict__ ele,
                                              const float* __restrict__ fc_w1,
                                              const int* __restrict__ edge_index,
                                              const float* __restrict__ BA,
                                              const float* __restrict__ BB,
                                              const float* __restrict__ BC,
                                              float* __restrict__ agg_s,
                                              float* __restrict__ agg_v) {
    __shared__ float sh_w1[64 * 16];
    __shared__ float sh_h[4][16][16];
    __shared__ float sh_s1[4][16][40];
    __shared__ float sh_b[4][16][20];
    __shared__ float sh_v1[4][16][3][20];
    __shared__ float sh_s2[4][16];
    __shared__ float sh_v2[4][16][3];
    __shared__ int sh_ni[4][16];

    if (blockIdx.x * 64 >= EE) return;  // uniform; grid is exact anyway

    const int tid = threadIdx.x;
    for (int t = tid; t < 1024; t += 128) sh_w1[t] = fc_w1[t];

    const int wv = tid >> 5;
    const int lane = tid & 31;
    const int m = lane & 15;
    const int half = lane >> 4;
    const int e = (blockIdx.x * 4 + wv) * 16 + m;

    const int ni = edge_index[e];
    const int nj = edge_index[EE + e];
    const float s2v = edge_sh[e * 4 + 0];
    const float v20 = edge_sh[e * 4 + 1];
    const float v21 = edge_sh[e * 4 + 2];
    const float v22 = edge_sh[e * 4 + 3];
    if (half == 0) {
        sh_ni[wv][m] = ni;
        sh_s2[wv][m] = s2v;
        sh_v2[wv][m][0] = v20; sh_v2[wv][m][1] = v21; sh_v2[wv][m][2] = v22;
    }
    // s1 = [s[i](16) | s[j](16) | es(8)]  -- this half writes 20 entries
    for (int t = 0; t < 20; t++) {
        int u = half * 20 + t;
        float val;
        if (u < 16)      val = s_pre[(size_t)ni * 16 + u];
        else if (u < 32) val = s_pre[(size_t)nj * 16 + (u - 16)];
        else             val = edge_fea[(size_t)e * 20 + (u - 32)];
        sh_s1[wv][m][u] = val;
    }
    // v1 = [v[i](8) | v[j](8) | ev(4)]  and  b[u] = RS3 * dot(v1[u], v2)
    for (int t = 0; t < 10; t++) {
        int u = half * 10 + t;
        float a0, a1, a2;
        if (u < 8) {
            const float* p = v_pre + (size_t)ni * 24 + u * 3;
            a0 = p[0]; a1 = p[1]; a2 = p[2];
        } else if (u < 16) {
            const float* p = v_pre + (size_t)nj * 24 + (u - 8) * 3;
            a0 = p[0]; a1 = p[1]; a2 = p[2];
        } else {
            const float* p = edge_fea + (size_t)e * 20 + 8 + (u - 16) * 3;
            a0 = p[0]; a1 = p[1]; a2 = p[2];
        }
        sh_v1[wv][m][0][u] = a0;
        sh_v1[wv][m][1][u] = a1;
        sh_v1[wv][m][2][u] = a2;
        sh_b[wv][m][u] = RS3f * (a0 * v20 + a1 * v21 + a2 * v22);
    }
    __syncthreads();  // sh_w1 ready
    // h: this lane computes 8 of the 16 hidden values of its edge
    {
        float hac[8];
#pragma unroll
        for (int t = 0; t < 8; t++) hac[t] = 0.0f;
        const float* ep = ele + (size_t)e * 64;
        for (int kk = 0; kk < 64; kk++) {
            float evl = ep[kk];
            const float* wr = &sh_w1[kk * 16 + half * 8];
#pragma unroll
            for (int t = 0; t < 8; t++) hac[t] += evl * wr[t];
        }
#pragma unroll
        for (int t = 0; t < 8; t++) sh_h[wv][m][half * 8 + t] = silu_f(hac[t] * 0.125f);
    }
    __syncthreads();  // all tile data ready

    const float* rs1 = sh_s1[wv][m];
    const float* rb = sh_b[wv][m];
    const float* rh = sh_h[wv][m];

    v8f z = {0, 0, 0, 0, 0, 0, 0, 0};
    v8f accA0 = z, accA1 = z, accB0 = z, accB1 = z, accC0 = z, accC1 = z, accC2 = z;

    // ---- G_A : A = h (x) s1, K = 640 ----
    for (int kh = 0; kh < 16; kh++) {
        float hm = rh[kh];
#pragma unroll
        for (int us = 0; us < 10; us++) {
            int ub = us * 4 + 2 * half;
            v2f a; a.x = hm * rs1[ub]; a.y = hm * rs1[ub + 1];
            int kstep = kh * 10 + us;
            const float* bp = BA + (size_t)kstep * 128 + lane * 2;
            v2f b0; b0.x = bp[0];  b0.y = bp[1];
            v2f b1; b1.x = bp[64]; b1.y = bp[65];
            accA0 = WMMA4(a, b0, accA0);
            accA1 = WMMA4(a, b1, accA1);
        }
    }
    // ---- G_B : A = h (x) b, K = 320 ----
    for (int kh = 0; kh < 16; kh++) {
        float hm = rh[kh];
#pragma unroll
        for (int us = 0; us < 5; us++) {
            int ub = us * 4 + 2 * half;
            v2f a; a.x = hm * rb[ub]; a.y = hm * rb[ub + 1];
            int kstep = kh * 5 + us;
            const float* bp = BB + (size_t)kstep * 128 + lane * 2;
            v2f b0; b0.x = bp[0];  b0.y = bp[1];
            v2f b1; b1.x = bp[64]; b1.y = bp[65];
            accB0 = WMMA4(a, b0, accB0);
            accB1 = WMMA4(a, b1, accB1);
        }
    }
    // ---- G_C : A = h (x) v1[:,:,c], K = 320, one 16-col tile ----
#pragma unroll
    for (int c = 0; c < 3; c++) {
        const float* rv = sh_v1[wv][m][c];
        v8f acc = z;
        for (int kh = 0; kh < 16; kh++) {
            float hm = rh[kh];
#pragma unroll
            for (int us = 0; us < 5; us++) {
                int ub = us * 4 + 2 * half;
                v2f a; a.x = hm * rv[ub]; a.y = hm * rv[ub + 1];
                int kstep = kh * 5 + us;
                const float* bp = BC + (size_t)kstep * 64 + lane * 2;
                v2f b0; b0.x = bp[0]; b0.y = bp[1];
                acc = WMMA4(a, b0, acc);
            }
        }
        if (c == 0) accC0 = acc;
        else if (c == 1) accC1 = acc;
        else accC2 = acc;
    }

    // ---- scatter: lane holds column n, rows r+8*(lane/16) ----
    const int n = lane & 15;
    const int rhf = lane >> 4;
#pragma unroll
    for (int r = 0; r < 8; r++) {
        int mm = r + 8 * rhf;
        int node = sh_ni[wv][mm];
        float s2 = sh_s2[wv][mm];
        float* as = agg_s + (size_t)node * 24;
        float* av = agg_v + (size_t)node * 24;
        atomicAdd(&as[n], s2 * accA0[r] + accB0[r]);
        if (n < 8) {
            atomicAdd(&as[16 + n], s2 * accA1[r] + accB1[r]);
            atomicAdd(&av[n * 3 + 0], s2 * accC0[r]);
            atomicAdd(&av[n * 3 + 1], s2 * accC1[r]);
            atomicAdd(&av[n * 3 + 2], s2 * accC2[r]);
        } else {
            int w = n - 8;
            float t3 = accA1[r];
            atomicAdd(&av[w * 3 + 0], t3 * sh_v2[wv][mm][0]);
            atomicAdd(&av[w * 3 + 1], t3 * sh_v2[wv][mm][1]);
            atomicAdd(&av[w * 3 + 2], t3 * sh_v2[wv][mm][2]);
        }
    }
}

// =====================================================================
// K3a: BN statistics. stats[0:24]=col sum, [24:48]=col sumsq, [48:56]=mean ||v||^2 sums
// =====================================================================
__global__ __launch_bounds__(256) void k3a_stats(const float* __restrict__ agg_s,
                                                 const float* __restrict__ agg_v,
                                                 float* __restrict__ stats) {
    __shared__ float ls[56];
    if (threadIdx.x < 56) ls[threadIdx.x] = 0.0f;
    __syncthreads();
    float acc[56];
#pragma unroll
    for (int t = 0; t < 56; t++) acc[t] = 0.0f;
    for (int nid = blockIdx.x * blockDim.x + threadIdx.x; nid < NN;
         nid += gridDim.x * blockDim.x) {
        const float* as = agg_s + (size_t)nid * 24;
#pragma unroll
        for (int w = 0; w < 24; w++) {
            float v = as[w];
            acc[w] += v;
            acc[24 + w] += v * v;
        }
        const float* av = agg_v + (size_t)nid * 24;
#pragma unroll
        for (int w = 0; w < 8; w++) {
            float v0 = av[w * 3], v1 = av[w * 3 + 1], v2 = av[w * 3 + 2];
            acc[48 + w] += v0 * v0 + v1 * v1 + v2 * v2;
        }
    }
#pragma unroll
    for (int t = 0; t < 56; t++) atomicAdd(&ls[t], acc[t]);
    __syncthreads();
    if (threadIdx.x < 56) atomicAdd(&stats[threadIdx.x], ls[threadIdx.x]);
}

// =====================================================================
// K3b: batchnorm + post-linear + gated output.
// =====================================================================
__global__ __launch_bounds__(256) void k3b_out(const float* __restrict__ agg_s,
                                               const float* __restrict__ agg_v,
                                               const float* __restrict__ stats,
                                               const float* __restrict__ bn_w,
                                               const float* __restrict__ bn_b,
                                               const float* __restrict__ Wps,
                                               const float* __restrict__ Wpv,
                                               float* __restrict__ out) {
    __shared__ float sWs[576];
    __shared__ float sWv[64];
    for (int t = threadIdx.x; t < 576; t += 256) sWs[t] = Wps[t];
    for (int t = threadIdx.x; t < 64; t += 256) sWv[t] = Wpv[t];
    __syncthreads();
    int nid = blockIdx.x * 256 + threadIdx.x;
    if (nid >= NN) return;
    const float invN = 1.0f / (float)NN;
    const float eps = 1e-5f;

    float bs[24];
#pragma unroll
    for (int w = 0; w < 24; w++) {
        float mu = stats[w] * invN;
        float var = stats[24 + w] * invN - mu * mu;
        bs[w] = (agg_s[(size_t)nid * 24 + w] - mu) * rsqrtf(var + eps) * bn_w[w] + bn_b[w];
    }
    float ps[24];
#pragma unroll
    for (int w = 0; w < 24; w++) ps[w] = 0.0f;
#pragma unroll
    for (int u = 0; u < 24; u++) {
        float b = bs[u];
        const float* wr = &sWs[u * 24];
#pragma unroll
        for (int w = 0; w < 24; w++) ps[w] += b * wr[w];
    }
#pragma unroll
    for (int w = 0; w < 24; w++) ps[w] *= RSQ24f;

    float* orow = out + (size_t)nid * 40;
#pragma unroll
    for (int w = 0; w < 16; w++) orow[w] = silu_f(ps[w]);
    float g[8];
#pragma unroll
    for (int w = 0; w < 8; w++) g[w] = sigmoid_f(ps[16 + w]);

    float bvn[8];
#pragma unroll
    for (int u = 0; u < 8; u++) bvn[u] = rsqrtf(stats[48 + u] * invN + eps) * bn_w[24 + u];
#pragma unroll
    for (int c = 0; c < 3; c++) {
        float pv[8];
#pragma unroll
        for (int w = 0; w < 8; w++) pv[w] = 0.0f;
#pragma unroll
        for (int u = 0; u < 8; u++) {
            float b = agg_v[(size_t)nid * 24 + u * 3 + c] * bvn[u];
            const float* wr = &sWv[u * 8];
#pragma unroll
            for (int w = 0; w < 8; w++) pv[w] += b * wr[w];
        }
#pragma unroll
        for (int w = 0; w < 8; w++) orow[16 + w * 3 + c] = pv[w] * RSQ8f * g[w];
    }
}

// =====================================================================
extern "C" void kernel_launch(void* const* d_in, const int* in_sizes, int n_in,
                              void* d_out, int out_size, void* d_ws, size_t ws_size,
                              hipStream_t stream) {
    const float* node_fea = (const float*)d_in[0];
    const float* edge_sh  = (const float*)d_in[1];
    const float* edge_fea = (const float*)d_in[2];
    const float* ele      = (const float*)d_in[3];
    const float* Wps      = (const float*)d_in[4];
    const float* Wpv      = (const float*)d_in[5];
    const float* Wss      = (const float*)d_in[6];
    const float* Wvv      = (const float*)d_in[7];
    const float* Wsv      = (const float*)d_in[8];
    const float* Wvs      = (const float*)d_in[9];
    const float* fc_w1    = (const float*)d_in[10];
    const float* fc_w2    = (const float*)d_in[11];
    const float* bn_w     = (const float*)d_in[12];
    const float* bn_b     = (const float*)d_in[13];
    const float* Wpost_s  = (const float*)d_in[14];
    const float* Wpost_v  = (const float*)d_in[15];
    const int* eidx       = (const int*)d_in[16];

    float* ws    = (float*)d_ws;
    float* s_pre = ws + OFF_SPRE;
    float* v_pre = ws + OFF_VPRE;
    float* agg_s = ws + OFF_AGGS;
    float* agg_v = ws + OFF_AGGV;
    float* BA    = ws + OFF_BA;
    float* BB    = ws + OFF_BB;
    float* BC    = ws + OFF_BC;
    float* stats = ws + OFF_STAT;

    k0_prep<<<(35904 + 255) / 256, 256, 0, stream>>>(fc_w2, BA, BB, BC, stats);
    k1_node<<<(NN + 255) / 256, 256, 0, stream>>>(node_fea, Wps, Wpv, Wss, Wvv, Wsv, Wvs,
                                                  s_pre, v_pre, agg_s, agg_v);
    k_edge<<<EE / 64, 128, 0, stream>>>(s_pre, v_pre, edge_sh, edge_fea, ele, fc_w1, eidx,
                                        BA, BB, BC, agg_s, agg_v);
    k3a_stats<<<120, 256, 0, stream>>>(agg_s, agg_v, stats);
    k3b_out<<<(NN + 255) / 256, 256, 0, stream>>>(agg_s, agg_v, stats, bn_w, bn_b, Wpost_s,
                                                  Wpost_v, (float*)d_out);
}